// GAT_62955630625071
// MI455X (gfx1250) — compile-verified
//
#include <hip/hip_runtime.h>
#include <hip/hip_bf16.h>

#define IN_CH   128
#define HID     32
#define OUT_CH  64
#define HEADS   4
#define NEG_SLOPE 0.2f

typedef float v2f __attribute__((ext_vector_type(2)));
typedef float v8f __attribute__((ext_vector_type(8)));

// ---------------------------------------------------------------------------
// WMMA fp32 GEMM: C[M x NC] = A[M x K] * B[K x NC].  K, NC compile-time.
// grid.x = ceil(M/16); blockDim.x = 32 * (NC/16)  (one wave per 16-col tile)
// Uses V_WMMA_F32_16X16X4_F32; A tile staged in LDS, shared by all waves;
// full K unroll so B addresses become immediate offsets.
// ---------------------------------------------------------------------------
template <int K, int NC>
__global__ __launch_bounds__(256) void gemm_wmma_f32(
    const float* __restrict__ A, const float* __restrict__ B,
    float* __restrict__ C, int M) {
  __shared__ float As[16 * K];
  const int row0 = blockIdx.x * 16;
  for (int i = threadIdx.x; i < 16 * K; i += blockDim.x) {
    const int r = i / K, c = i % K;
    const int row = row0 + r;
    As[i] = (row < M) ? A[(size_t)row * K + c] : 0.0f;
  }
  __syncthreads();

  const int lane = threadIdx.x & 31;
  const int wave = threadIdx.x >> 5;
  const int col0 = wave * 16;
  const int half = lane >> 4;   // ISA 7.12.2: lanes 0-15 hold K=0,1 / lanes 16-31 hold K=2,3
  const int rl   = lane & 15;

  const float* Bp = B + col0 + rl;   // single base pointer; rest immediate offsets
  const float* Ap = As + rl * K;

  v8f acc = {};
#pragma unroll
  for (int k0 = 0; k0 < K; k0 += 4) {
    const int ka = k0 + half * 2;
    v2f a, b;
    a.x = Ap[ka];                  // A: M=rl, K=ka
    a.y = Ap[ka + 1];              // A: M=rl, K=ka+1
    b.x = Bp[(size_t)ka * NC];     // B: K=ka,   N=rl
    b.y = Bp[(size_t)(ka + 1) * NC];
    acc = __builtin_amdgcn_wmma_f32_16x16x4_f32(false, a, false, b,
                                                (short)0, acc, false, false);
  }

  // D layout: VGPR r -> M = r + 8*half (rows split at lane 16), N = rl
  const int rowb = row0 + half * 8;
#pragma unroll
  for (int r = 0; r < 8; ++r) {
    const int row = rowb + r;
    if (row < M) C[(size_t)row * NC + col0 + rl] = acc[r];
  }
}

// ---------------------------------------------------------------------------
// a_s[n,h] = sum_c h[n,h,c]*att_src[h,c] ; a_d likewise. One thread per (n,h).
// ---------------------------------------------------------------------------
template <int H, int C>
__global__ void att_dots(const float* __restrict__ h,
                         const float* __restrict__ att_s,
                         const float* __restrict__ att_d,
                         float* __restrict__ a_s, float* __restrict__ a_d,
                         int N) {
  const long long t = (long long)blockIdx.x * blockDim.x + threadIdx.x;
  if (t >= (long long)N * H) return;
  const int n = (int)(t / H), hh = (int)(t % H);
  const float* row = h + (size_t)n * H * C + (size_t)hh * C;
  const float* ws  = att_s + (size_t)hh * C;
  const float* wd  = att_d + (size_t)hh * C;
  float ss = 0.f, sd = 0.f;
#pragma unroll 8
  for (int c = 0; c < C; ++c) { ss += row[c] * ws[c]; sd += row[c] * wd[c]; }
  a_s[t] = ss; a_d[t] = sd;
}

__global__ void fill_f32(float* __restrict__ p, long long n, float v) {
  const long long t = (long long)blockIdx.x * blockDim.x + threadIdx.x;
  if (t < n) p[t] = v;
}

__device__ __forceinline__ void edge_sd(const int* __restrict__ ei, int E,
                                        long long e, int& s, int& d) {
  if (e < (long long)E) { s = ei[e]; d = ei[(size_t)E + e]; }
  else                  { s = d = (int)(e - E); }   // self-loop
}

// Race-free float atomic max via signed/unsigned integer monotonicity trick.
__device__ __forceinline__ void atomic_max_f32(float* addr, float val) {
  if (val >= 0.0f) atomicMax((int*)addr, __float_as_int(val));
  else             atomicMin((unsigned int*)addr, (unsigned int)__float_as_int(val));
}

// Pass 1: segment max of leaky_relu(a_s[src]+a_d[dst]) over dst. (edge,head) thread.
template <int H>
__global__ void edge_lmax(const int* __restrict__ ei, int E, int N,
                          const float* __restrict__ a_s,
                          const float* __restrict__ a_d,
                          float* __restrict__ m) {
  const long long t = (long long)blockIdx.x * blockDim.x + threadIdx.x;
  const long long tot = (long long)(E + N) * H;
  if (t >= tot) return;
  const int hh = (int)(t & (H - 1));
  const long long e = t >> __builtin_ctz(H);
  int s, d; edge_sd(ei, E, e, s, d);
  float el = a_s[(size_t)s * H + hh] + a_d[(size_t)d * H + hh];
  el = (el > 0.f) ? el : NEG_SLOPE * el;
  atomic_max_f32(&m[(size_t)d * H + hh], el);
}

// Pass 2: e_exp = exp(e - m[dst]); denom[dst] += e_exp. (edge,head) thread.
template <int H>
__global__ void edge_expsum(const int* __restrict__ ei, int E, int N,
                            const float* __restrict__ a_s,
                            const float* __restrict__ a_d,
                            const float* __restrict__ m,
                            float* __restrict__ eexp,
                            float* __restrict__ denom) {
  const long long t = (long long)blockIdx.x * blockDim.x + threadIdx.x;
  const long long tot = (long long)(E + N) * H;
  if (t >= tot) return;
  const int hh = (int)(t & (H - 1));
  const long long e = t >> __builtin_ctz(H);
  int s, d; edge_sd(ei, E, e, s, d);
  float el = a_s[(size_t)s * H + hh] + a_d[(size_t)d * H + hh];
  el = (el > 0.f) ? el : NEG_SLOPE * el;
  const float ex = __expf(el - m[(size_t)d * H + hh]);
  eexp[t] = ex;
  atomicAdd(&denom[(size_t)d * H + hh], ex);
}

// Pass 3: out[dst,ch] += h[src,ch] * e_exp/(denom[dst]+eps). (edge,channel) thread.
template <int H, int C>
__global__ void edge_scatter(const int* __restrict__ ei, int E, int N,
                             const float* __restrict__ hfeat,
                             const float* __restrict__ eexp,
                             const float* __restrict__ denom,
                             float* __restrict__ out) {
  constexpr int HC = H * C;
  const long long t = (long long)blockIdx.x * blockDim.x + threadIdx.x;
  const long long tot = (long long)(E + N) * HC;
  if (t >= tot) return;
  const int ch = (int)(t & (HC - 1));        // HC power of two
  const long long e = t >> __builtin_ctz(HC);
  const int hh = ch / C;                      // C power of two -> shift
  int s, d; edge_sd(ei, E, e, s, d);
  const float alpha = eexp[e * H + hh] / (denom[(size_t)d * H + hh] + 1e-16f);
  atomicAdd(&out[(size_t)d * HC + ch], hfeat[(size_t)s * HC + ch] * alpha);
}

__global__ void bias_elu(float* __restrict__ p, const float* __restrict__ b,
                         long long n, int HC) {
  const long long t = (long long)blockIdx.x * blockDim.x + threadIdx.x;
  if (t >= n) return;
  const float v = p[t] + b[t & (HC - 1)];
  p[t] = (v > 0.f) ? v : (__expf(v) - 1.0f);
}

__global__ void bias_add(float* __restrict__ p, const float* __restrict__ b,
                         long long n, int HC) {
  const long long t = (long long)blockIdx.x * blockDim.x + threadIdx.x;
  if (t >= n) return;
  p[t] += b[t & (HC - 1)];
}

// ---------------------------------------------------------------------------
extern "C" void kernel_launch(void* const* d_in, const int* in_sizes, int n_in,
                              void* d_out, int out_size, void* d_ws, size_t ws_size,
                              hipStream_t stream) {
  (void)n_in; (void)out_size; (void)ws_size;
  const float* x   = (const float*)d_in[0];
  const int*   ei  = (const int*)d_in[1];      // [2, E]
  const float* W1  = (const float*)d_in[2];    // [128, 128]
  const float* as1 = (const float*)d_in[3];    // [4, 32]
  const float* ad1 = (const float*)d_in[4];
  const float* b1  = (const float*)d_in[5];    // [128]
  const float* W2  = (const float*)d_in[6];    // [128, 64]
  const float* as2 = (const float*)d_in[7];    // [1, 64]
  const float* ad2 = (const float*)d_in[8];
  const float* b2  = (const float*)d_in[9];    // [64]

  const int N = in_sizes[0] / IN_CH;
  const int E = in_sizes[1] / 2;
  const long long ET = (long long)E + N;       // edges incl. self-loops

  char* wsp = (char*)d_ws;
  auto carve = [&](size_t bytes) -> float* {
    char* p = wsp; wsp += (bytes + 255) & ~(size_t)255; return (float*)p;
  };
  float* h1  = carve((size_t)N * 128 * sizeof(float));     // x @ W1
  float* aS1 = carve((size_t)N * HEADS * sizeof(float));
  float* aD1 = carve((size_t)N * HEADS * sizeof(float));
  float* m1  = carve((size_t)N * HEADS * sizeof(float));
  float* dn1 = carve((size_t)N * HEADS * sizeof(float));
  float* ex1 = carve((size_t)ET * HEADS * sizeof(float));
  float* o1  = carve((size_t)N * 128 * sizeof(float));     // layer-1 out -> ELU'd features
  float* h2  = carve((size_t)N * OUT_CH * sizeof(float));  // o1 @ W2
  float* aS2 = carve((size_t)N * sizeof(float));
  float* aD2 = carve((size_t)N * sizeof(float));
  float* m2  = carve((size_t)N * sizeof(float));
  float* dn2 = carve((size_t)N * sizeof(float));
  float* ex2 = carve((size_t)ET * sizeof(float));
  float* out = (float*)d_out;                              // [N, 64]

  auto blocks = [](long long n, int bs) -> unsigned {
    return (unsigned)((n + bs - 1) / bs);
  };
  const float NEG_INF = -__builtin_huge_valf();

  // ---------------- Layer 1 (H=4, C=32, concat) ----------------
  gemm_wmma_f32<128, 128><<<blocks(N, 16), 256, 0, stream>>>(x, W1, h1, N);
  att_dots<HEADS, HID><<<blocks((long long)N * HEADS, 256), 256, 0, stream>>>(
      h1, as1, ad1, aS1, aD1, N);
  fill_f32<<<blocks((long long)N * HEADS, 256), 256, 0, stream>>>(
      m1, (long long)N * HEADS, NEG_INF);
  fill_f32<<<blocks((long long)N * HEADS, 256), 256, 0, stream>>>(
      dn1, (long long)N * HEADS, 0.0f);
  fill_f32<<<blocks((long long)N * 128, 256), 256, 0, stream>>>(
      o1, (long long)N * 128, 0.0f);
  edge_lmax<HEADS><<<blocks(ET * HEADS, 256), 256, 0, stream>>>(ei, E, N, aS1, aD1, m1);
  edge_expsum<HEADS><<<blocks(ET * HEADS, 256), 256, 0, stream>>>(
      ei, E, N, aS1, aD1, m1, ex1, dn1);
  edge_scatter<HEADS, HID><<<blocks(ET * 128, 256), 256, 0, stream>>>(
      ei, E, N, h1, ex1, dn1, o1);
  bias_elu<<<blocks((long long)N * 128, 256), 256, 0, stream>>>(
      o1, b1, (long long)N * 128, 128);

  // ---------------- Layer 2 (H=1, C=64, mean == identity) ----------------
  gemm_wmma_f32<128, OUT_CH><<<blocks(N, 16), 128, 0, stream>>>(o1, W2, h2, N);
  att_dots<1, OUT_CH><<<blocks(N, 256), 256, 0, stream>>>(h2, as2, ad2, aS2, aD2, N);
  fill_f32<<<blocks(N, 256), 256, 0, stream>>>(m2, (long long)N, NEG_INF);
  fill_f32<<<blocks(N, 256), 256, 0, stream>>>(dn2, (long long)N, 0.0f);
  fill_f32<<<blocks((long long)N * OUT_CH, 256), 256, 0, stream>>>(
      out, (long long)N * OUT_CH, 0.0f);
  edge_lmax<1><<<blocks(ET, 256), 256, 0, stream>>>(ei, E, N, aS2, aD2, m2);
  edge_expsum<1><<<blocks(ET, 256), 256, 0, stream>>>(ei, E, N, aS2, aD2, m2, ex2, dn2);
  edge_scatter<1, OUT_CH><<<blocks(ET * OUT_CH, 256), 256, 0, stream>>>(
      ei, E, N, h2, ex2, dn2, out);
  bias_add<<<blocks((long long)N * OUT_CH, 256), 256, 0, stream>>>(
      out, b2, (long long)N * OUT_CH, OUT_CH);
}